// FP4Quantizer_5987184411061
// MI455X (gfx1250) — compile-verified
//
#include <hip/hip_runtime.h>
#include <stdint.h>

// FP4-E2M1 per-row fake-quant, single-pass, HBM-bandwidth-bound.
// Row (16KB) staged via CDNA5 async global->LDS DMA (ASYNCcnt path).
// Quantization done with a 1-mantissa-bit float rounding bit-trick
// (~10 VALU/element) so the kernel stays under the 23.3 TB/s roofline.

#define ROW_LEN        4096
#define BLOCK_THREADS  256
#define VEC4_PER_ROW   (ROW_LEN / 4)                 // 1024 float4 per row
#define CHUNKS         (VEC4_PER_ROW / BLOCK_THREADS) // 4 float4 per thread

typedef int v4i __attribute__((vector_size(16)));
typedef __attribute__((address_space(1))) v4i GV4i;   // global v4i32
typedef __attribute__((address_space(3))) v4i LV4i;   // LDS v4i32
typedef __attribute__((address_space(3))) void LVoid;

__device__ __forceinline__ void async_load_b128(const float* gsrc, float* lds_dst) {
#if __has_builtin(__builtin_amdgcn_global_load_async_to_lds_b128)
  __builtin_amdgcn_global_load_async_to_lds_b128(
      (GV4i*)(void*)gsrc, (LV4i*)(void*)lds_dst, /*offset=*/0, /*cpol=*/0);
#else
  unsigned lds_off = (unsigned)(uintptr_t)(LVoid*)(void*)lds_dst;
  asm volatile("global_load_async_to_lds_b128 %0, %1, off"
               :
               : "v"(lds_off), "v"(gsrc)
               : "memory");
#endif
}

__device__ __forceinline__ void wait_async_zero() {
#if __has_builtin(__builtin_amdgcn_s_wait_asynccnt)
  __builtin_amdgcn_s_wait_asynccnt(0);
#else
  asm volatile("s_wait_asynccnt 0" ::: "memory");
#endif
}

// Wave32 XOR-butterfly max via ds_swizzle (group-of-32 mode:
// offset = {0, and=0x1F [14:10], or=0 [9:5], xor [4:0]} = 0x7C00 | xor).
// Immediate must be an ICE at the call site -> template parameter.
template <int IMM>
__device__ __forceinline__ float swz_max(float m) {
  float o = __int_as_float(__builtin_amdgcn_ds_swizzle(__float_as_int(m), IMM));
  return fmaxf(m, o);
}

// Fake-quant one element: x -> nearest codebook level * s.
// Codebook {0, ±m*2^e : e in [-2,1], m in {1,1.5}} == round-to-1-mantissa-bit
// in [0.25, 2]; thresholds 1.25*2^e / 1.75*2^e match the argmin midpoints.
// (Exact-midpoint tie direction differs from jnp.argmin; measure-zero.)
__device__ __forceinline__ float fp4_fakequant(float x, float inv_s, float s) {
  const float a = fabsf(x) * inv_s;                      // v_mul (|x| src mod)
  unsigned r = (__float_as_uint(a) + 0x00200000u) & 0xFFC00000u;
  float q = __uint_as_float(r);                          // 1-mantissa-bit round
  q = fminf(q, 2.0f);                                    // top level cap
  q = fmaxf(q, 0.25f);                                   // bottom nonzero level
  q = (a > 0.125f) ? q : 0.0f;                           // zero region
  return __builtin_copysignf(q * s, x);                  // v_bfi_b32
}

__global__ __launch_bounds__(BLOCK_THREADS)
void fp4_quant_rows(const float* __restrict__ x, float* __restrict__ out) {
  __shared__ __align__(16) float smem[ROW_LEN];
  __shared__ float wave_max[BLOCK_THREADS / 32];

  const int tid = threadIdx.x;
  const int row = blockIdx.x;
  const float* xr   = x   + (size_t)row * ROW_LEN;
  float*       outr = out + (size_t)row * ROW_LEN;

  // ---- Phase 1: async DMA the whole row into LDS (ASYNCcnt-tracked) ----
#pragma unroll
  for (int c = 0; c < CHUNKS; ++c) {
    const int u = c * BLOCK_THREADS + tid;      // float4 index in row
    async_load_b128(xr + u * 4, &smem[u * 4]);
  }
  wait_async_zero();     // this wave's DMA chunks are in LDS
  __syncthreads();       // publish across the 8 waves

  // ---- Phase 2: per-thread |max| over its 16 elements (LDS -> regs) ----
  float4 v[CHUNKS];
  float m = 0.0f;
#pragma unroll
  for (int c = 0; c < CHUNKS; ++c) {
    const int u = c * BLOCK_THREADS + tid;
    v[c] = *reinterpret_cast<const float4*>(&smem[u * 4]);  // ds_load_b128
    m = fmaxf(m, fabsf(v[c].x));   // folds to v_max_f32 with |src|
    m = fmaxf(m, fabsf(v[c].y));
    m = fmaxf(m, fabsf(v[c].z));
    m = fmaxf(m, fabsf(v[c].w));
  }
  // wave32 butterfly max via ds_swizzle (immediate offsets)
  m = swz_max<0x7C00 | 16>(m);
  m = swz_max<0x7C00 | 8>(m);
  m = swz_max<0x7C00 | 4>(m);
  m = swz_max<0x7C00 | 2>(m);
  m = swz_max<0x7C00 | 1>(m);
  if ((tid & 31) == 0) wave_max[tid >> 5] = m;
  __syncthreads();
  float rowmax = wave_max[0];
#pragma unroll
  for (int w = 1; w < BLOCK_THREADS / 32; ++w)
    rowmax = fmaxf(rowmax, wave_max[w]);

  // s = clamp(maxabs / (2.0 + 1e-12), 1e-8, inf); (2.0f + 1e-12f == 2.0f)
  const float maxabs = fmaxf(rowmax, 1e-12f);
  const float s      = fmaxf(maxabs * 0.5f, 1e-8f);
  const float inv_s  = 1.0f / s;

  // ---- Phase 3: quantize + vector store ----
#pragma unroll
  for (int c = 0; c < CHUNKS; ++c) {
    const int u = c * BLOCK_THREADS + tid;
    float4 q;
    q.x = fp4_fakequant(v[c].x, inv_s, s);
    q.y = fp4_fakequant(v[c].y, inv_s, s);
    q.z = fp4_fakequant(v[c].z, inv_s, s);
    q.w = fp4_fakequant(v[c].w, inv_s, s);
    *reinterpret_cast<float4*>(outr + u * 4) = q;           // global_store_b128
  }
}

extern "C" void kernel_launch(void* const* d_in, const int* in_sizes, int n_in,
                              void* d_out, int out_size, void* d_ws, size_t ws_size,
                              hipStream_t stream) {
  (void)n_in; (void)d_ws; (void)ws_size; (void)out_size;
  const float* x = (const float*)d_in[0];   // [rows, 4096] fp32
  // d_in[1] (levels, 15 fp32) is a fixed codebook; baked into the kernel.
  float* out = (float*)d_out;

  const int total = in_sizes[0];
  const int rows  = total / ROW_LEN;        // 4096 for the reference shape

  fp4_quant_rows<<<dim3(rows), dim3(BLOCK_THREADS), 0, stream>>>(x, out);
}